// RTE_26834955665767
// MI455X (gfx1250) — compile-verified
//
#include <hip/hip_runtime.h>
#include <hip/hip_bf16.h>
#include <math.h>

#define BB 16
#define NN 2304
#define DD 1024
#define HH 1344
#define WWi 1344
#define MM 8
#define HV 48
#define PH 28
#define NPIX (HH*WWi)
#define NRAYS 24
#define NSAMP 20
#define KSTEEP 50.0f
#define THETA 0.3f
#define MAXR 604.8f            /* min(H,W)*0.45 */
#define NCHUNK 4
#define NPERCHUNK (NN/NCHUNK)  /* 576 */
#define RGB_ELEMS (MM*BB*DD)   /* 131072 */

typedef float v2f __attribute__((ext_vector_type(2)));
typedef float v8f __attribute__((ext_vector_type(8)));

__device__ __forceinline__ float warpSum(float v) {
  #pragma unroll
  for (int o = 16; o; o >>= 1) v += __shfl_xor(v, o, 32);
  return v;
}
__device__ __forceinline__ float warpMax(float v) {
  #pragma unroll
  for (int o = 16; o; o >>= 1) v = fmaxf(v, __shfl_xor(v, o, 32));
  return v;
}
__device__ __forceinline__ float blockSum(float v, float* sm) {
  __syncthreads();
  v = warpSum(v);
  int lane = threadIdx.x & 31, w = threadIdx.x >> 5, nw = (blockDim.x + 31) >> 5;
  if (lane == 0) sm[w] = v;
  __syncthreads();
  if (w == 0) {
    float x = (lane < nw) ? sm[lane] : 0.f;
    x = warpSum(x);
    if (lane == 0) sm[0] = x;
  }
  __syncthreads();
  return sm[0];
}
__device__ __forceinline__ float blockMax(float v, float* sm) {
  __syncthreads();
  v = warpMax(v);
  int lane = threadIdx.x & 31, w = threadIdx.x >> 5, nw = (blockDim.x + 31) >> 5;
  if (lane == 0) sm[w] = v;
  __syncthreads();
  if (w == 0) {
    float x = (lane < nw) ? sm[lane] : -3.0e38f;
    x = warpMax(x);
    if (lane == 0) sm[0] = x;
  }
  __syncthreads();
  return sm[0];
}

// ---------------- scores[b,n] = <vt[b,n,:], gate_w> : one wave per row -------
__global__ void __launch_bounds__(256) k_scores(const float* __restrict__ vt,
                                                const float* __restrict__ gw,
                                                float* __restrict__ scores) {
  int wave = threadIdx.x >> 5, lane = threadIdx.x & 31;
  int row = blockIdx.x * 8 + wave;                // 36864 rows = 4608 blocks * 8
  const float4* p = (const float4*)(vt + (size_t)row * DD);
  const float4* g = (const float4*)gw;
  float acc = 0.f;
  #pragma unroll
  for (int k = 0; k < 8; ++k) {                   // 8*32*4 = 1024
    float4 a = p[k * 32 + lane];
    float4 w = g[k * 32 + lane];
    acc += a.x * w.x + a.y * w.y + a.z * w.z + a.w * w.w;
  }
  acc = warpSum(acc);
  if (lane == 0) scores[row] = acc;
}

// ------- coverage cell sums for all 8 masks + bit-packed mask byte ----------
__global__ void __launch_bounds__(256) k_cov(const int* __restrict__ masks,
                                             int* __restrict__ cellsum,
                                             unsigned char* __restrict__ packed) {
  int cell = blockIdx.x;                          // 2304 cells
  int cy = cell / HV, cx = cell % HV;
  int acc[MM];
  #pragma unroll
  for (int m = 0; m < MM; ++m) acc[m] = 0;
  for (int i = threadIdx.x; i < PH * PH; i += 256) {
    int py = cy * PH + i / PH, px = cx * PH + i % PH;
    int pix = py * WWi + px;
    unsigned int byte = 0;
    #pragma unroll
    for (int m = 0; m < MM; ++m) {
      int v = masks[(size_t)m * NPIX + pix];
      acc[m] += v;
      byte |= (unsigned)(v & 1) << m;
    }
    packed[pix] = (unsigned char)byte;
  }
  __shared__ int sm[MM][256];
  #pragma unroll
  for (int m = 0; m < MM; ++m) sm[m][threadIdx.x] = acc[m];
  __syncthreads();
  for (int s = 128; s > 0; s >>= 1) {
    if (threadIdx.x < s)
      #pragma unroll
      for (int m = 0; m < MM; ++m) sm[m][threadIdx.x] += sm[m][threadIdx.x + s];
    __syncthreads();
  }
  if (threadIdx.x < MM) cellsum[threadIdx.x * HV * HV + cell] = sm[threadIdx.x][0];
}

// -------- soft2d, log_soft, count, centroid per mask ------------------------
__global__ void __launch_bounds__(256) k_soft(const int* __restrict__ cellsum,
                                              float* __restrict__ soft2d,
                                              float* __restrict__ logsoft,
                                              float* __restrict__ scal) {
  int m = blockIdx.x;
  __shared__ float red[32];
  float cnt = 0.f, ssum = 0.f, sx = 0.f, sy = 0.f;
  for (int j = threadIdx.x; j < HV * HV; j += 256) {
    int cs = cellsum[m * HV * HV + j];
    float cov = (float)cs * (1.0f / (PH * PH));
    float s = 1.0f / (1.0f + expf(-KSTEEP * (cov - THETA)));
    soft2d[m * HV * HV + j] = s;
    logsoft[m * HV * HV + j] = logf(fmaxf(s, 1e-8f));
    cnt += (float)cs;
    ssum += s;
    sx += s * (float)(j % HV);
    sy += s * (float)(j / HV);
  }
  cnt  = blockSum(cnt, red);
  ssum = blockSum(ssum, red);
  sx   = blockSum(sx, red);
  sy   = blockSum(sy, red);
  if (threadIdx.x == 0) {
    float se = ssum + 1e-8f;
    float cxn = sx / (se * (float)HV);
    float cyn = sy / (se * (float)HV);
    scal[m * 8 + 0] = cnt;
    scal[m * 8 + 1] = cxn;
    scal[m * 8 + 2] = cyn;
    scal[m * 8 + 3] = cxn * (float)(WWi - 1);
    scal[m * 8 + 4] = cyn * (float)(HH - 1);
  }
}

// ---- depth pass: per-block partial Σd, Σd² for all 8 masks (packed byte) ---
#define DGRID 256
__global__ void __launch_bounds__(256) k_depth(const float* __restrict__ depth,
                                               const unsigned char* __restrict__ packed,
                                               float* __restrict__ dpart) {
  int b = blockIdx.y;
  float sd[MM], sq[MM];
  #pragma unroll
  for (int m = 0; m < MM; ++m) { sd[m] = 0.f; sq[m] = 0.f; }
  const float* dp = depth + (size_t)b * NPIX;
  for (int p = blockIdx.x * 256 + threadIdx.x; p < NPIX; p += DGRID * 256) {
    float d = dp[p];
    unsigned pk = packed[p];
    float d2 = d * d;
    #pragma unroll
    for (int m = 0; m < MM; ++m) {
      float fm = (float)((pk >> m) & 1u);
      sd[m] += d * fm;
      sq[m] += d2 * fm;
    }
  }
  #pragma unroll
  for (int m = 0; m < MM; ++m) { sd[m] = warpSum(sd[m]); sq[m] = warpSum(sq[m]); }
  __shared__ float sm[8][16];
  int lane = threadIdx.x & 31, w = threadIdx.x >> 5;
  if (lane == 0) {
    #pragma unroll
    for (int m = 0; m < MM; ++m) { sm[w][m] = sd[m]; sm[w][8 + m] = sq[m]; }
  }
  __syncthreads();
  if (threadIdx.x < 16) {
    float a = 0.f;
    #pragma unroll
    for (int ww = 0; ww < 8; ++ww) a += sm[ww][threadIdx.x];
    dpart[((size_t)b * DGRID + blockIdx.x) * 16 + threadIdx.x] = a;
  }
}
// deterministic second-stage reduction -> sumd[m][b], sumd2[m][b]
__global__ void __launch_bounds__(256) k_depth2(const float* __restrict__ dpart,
                                                float* __restrict__ sumd,
                                                float* __restrict__ sumd2) {
  int b = blockIdx.x;
  int idx = threadIdx.x & 15, seg = threadIdx.x >> 4;   // 16 segs * 16 idx
  float a = 0.f;
  for (int j = seg; j < DGRID; j += 16)
    a += dpart[((size_t)b * DGRID + j) * 16 + idx];
  __shared__ float sm[16][17];
  sm[seg][idx] = a;
  __syncthreads();
  if (threadIdx.x < 16) {
    float t = 0.f;
    #pragma unroll
    for (int s = 0; s < 16; ++s) t += sm[s][threadIdx.x];
    int m = threadIdx.x & 7;
    if (threadIdx.x < 8) sumd[m * BB + b] = t;
    else                 sumd2[m * BB + b] = t;
  }
}

// -------- softmax over n per (m,b); write WMMA-A padded layout [b][n][16] ---
__global__ void __launch_bounds__(256) k_weights(const float* __restrict__ scores,
                                                 const float* __restrict__ logsoft,
                                                 float* __restrict__ wpad) {
  int b = blockIdx.x, m = blockIdx.y;
  __shared__ float red[32];
  float v[9];                                    // 2304 = 9 * 256 exactly
  float mx = -3.0e38f;
  #pragma unroll
  for (int k = 0; k < 9; ++k) {
    int n = threadIdx.x + k * 256;
    float x = scores[b * NN + n] + logsoft[m * NN + n];
    v[k] = x;
    mx = fmaxf(mx, x);
  }
  mx = blockMax(mx, red);
  float se = 0.f;
  #pragma unroll
  for (int k = 0; k < 9; ++k) { v[k] = expf(v[k] - mx); se += v[k]; }
  se = blockSum(se, red);
  float inv = 1.0f / se;
  #pragma unroll
  for (int k = 0; k < 9; ++k) {
    int n = threadIdx.x + k * 256;
    size_t base = ((size_t)b * NN + n) * 16;
    wpad[base + m] = v[k] * inv;
    if (m == 0) {                                // zero the pad rows 8..15 once
      #pragma unroll
      for (int j = 8; j < 16; ++j) wpad[base + j] = 0.f;
    }
  }
}

// -------- rgb GEMM via V_WMMA_F32_16X16X4_F32 --------------------------------
// A = weights (16 masks-padded x 4 tokens), B = vt (4 tokens x 16 dims)
__global__ void __launch_bounds__(256) k_rgb(const float* __restrict__ vt,
                                             const float* __restrict__ wpad,
                                             float* __restrict__ partial) {
  int wave = threadIdx.x >> 5, lane = threadIdx.x & 31;
  int b = blockIdx.y, chunk = blockIdx.z;
  int d0 = (blockIdx.x * 8 + wave) * 16;         // 8 blocks.x * 8 waves -> 64 d-tiles
  int r = lane & 15, hi = lane >> 4;
  int nbase = chunk * NPERCHUNK;
  const float* v0 = vt + ((size_t)b * NN + nbase + 2 * hi) * DD + d0 + r;
  const float* w0 = wpad + ((size_t)b * NN + nbase + 2 * hi) * 16 + r;
  v8f c = {0.f, 0.f, 0.f, 0.f, 0.f, 0.f, 0.f, 0.f};
  #pragma unroll 4
  for (int it = 0; it < NPERCHUNK / 4; ++it) {   // 144 WMMAs per wave
    v2f a, bf;
    a.x  = w0[0];   a.y  = w0[16];               // A: K=2hi, K=2hi+1 of row r
    bf.x = v0[0];   bf.y = v0[DD];               // B: rows K=2hi, K=2hi+1, col d0+r
    c = __builtin_amdgcn_wmma_f32_16x16x4_f32(false, a, false, bf,
                                              (short)0, c, false, false);
    w0 += 4 * 16;
    v0 += 4 * (size_t)DD;
  }
  if (hi == 0) {                                 // D rows M=0..7 live in hi=0 half
    float* o = partial + (size_t)chunk * RGB_ELEMS;
    #pragma unroll
    for (int v = 0; v < 8; ++v)
      o[((size_t)(v * BB + b)) * DD + d0 + r] = c[v];
  }
}
__global__ void __launch_bounds__(256) k_rgbsum(const float* __restrict__ partial,
                                                float* __restrict__ out) {
  int i = blockIdx.x * 256 + threadIdx.x;
  if (i < RGB_ELEMS) {
    float s = partial[i];
    #pragma unroll
    for (int cch = 1; cch < NCHUNK; ++cch) s += partial[(size_t)cch * RGB_ELEMS + i];
    out[i] = s;
  }
}

// -------- ray sampling + per-(m,b) stats vector (28) ------------------------
__global__ void __launch_bounds__(512) k_rays(const float* __restrict__ depth,
                                              const float* __restrict__ soft2d,
                                              const float* __restrict__ scal,
                                              const float* __restrict__ sumd,
                                              const float* __restrict__ sumd2,
                                              float* __restrict__ stats) {
  int b = blockIdx.x, m = blockIdx.y;
  __shared__ float sd_[NRAYS * NSAMP], sw_[NRAYS * NSAMP];
  float cnt = scal[m * 8 + 0];
  float cxn = scal[m * 8 + 1], cyn = scal[m * 8 + 2];
  float cxp = scal[m * 8 + 3], cyp = scal[m * 8 + 4];
  int t = threadIdx.x;
  if (t < NRAYS * NSAMP) {
    int ray = t / NSAMP, s = t % NSAMP;
    float ang = 6.283185307179586f * (float)ray / (float)NRAYS;
    float tv = (MAXR / (float)(NSAMP - 1)) * (float)s;
    float x = cxp + cosf(ang) * tv;
    float y = cyp + sinf(ang) * tv;
    // bilinear, border mode, on depth[b]
    float xc = fminf(fmaxf(x, 0.f), (float)(WWi - 1));
    float yc = fminf(fmaxf(y, 0.f), (float)(HH - 1));
    float x0f = floorf(xc), y0f = floorf(yc);
    float wx = xc - x0f, wy = yc - y0f;
    int x0 = (int)x0f, y0 = (int)y0f;
    int x1 = min(x0 + 1, WWi - 1), y1 = min(y0 + 1, HH - 1);
    const float* dp = depth + (size_t)b * NPIX;
    float v00 = dp[(size_t)y0 * WWi + x0], v01 = dp[(size_t)y0 * WWi + x1];
    float v10 = dp[(size_t)y1 * WWi + x0], v11 = dp[(size_t)y1 * WWi + x1];
    sd_[t] = v00 * (1.f - wx) * (1.f - wy) + v01 * wx * (1.f - wy)
           + v10 * (1.f - wx) * wy + v11 * wx * wy;
    // bilinear, zeros mode, on soft2d[m] (48x48), un-clipped coords
    float xp = x * (1.0f / (float)(WWi - 1)) * (float)(HV - 1);
    float yp = y * (1.0f / (float)(HH - 1)) * (float)(HV - 1);
    float px0f = floorf(xp), py0f = floorf(yp);
    float pwx = xp - px0f, pwy = yp - py0f;
    int px0 = (int)px0f, py0 = (int)py0f;
    const float* sp = soft2d + m * HV * HV;
    auto tap = [&](int yy, int xx, float w) -> float {
      bool valid = (xx >= 0) & (xx < HV) & (yy >= 0) & (yy < HV);
      int yyc = min(max(yy, 0), HV - 1), xxc = min(max(xx, 0), HV - 1);
      float vv = sp[yyc * HV + xxc];
      return valid ? vv * w : 0.f;
    };
    float sw = tap(py0, px0, (1.f - pwx) * (1.f - pwy))
             + tap(py0, px0 + 1, pwx * (1.f - pwy))
             + tap(py0 + 1, px0, (1.f - pwx) * pwy)
             + tap(py0 + 1, px0 + 1, pwx * pwy);
    sw_[t] = fmaxf(sw, 1e-6f);
  }
  __syncthreads();
  float gate = (cnt > 0.f) ? 1.f : 0.f;
  float* st = stats + ((size_t)(m * BB + b)) * 28;
  if (t < NRAYS) {
    float ssum = 0.f, pr = 0.f;
    #pragma unroll
    for (int s = 0; s < NSAMP; ++s) {
      float w = sw_[t * NSAMP + s];
      ssum += w;
      pr += sd_[t * NSAMP + s] * w;
    }
    st[4 + t] = (pr / ssum) * gate;
  } else if (t >= 32 && t < 36) {
    int which = t - 32;
    float s = sumd[m * BB + b], sq = sumd2[m * BB + b];
    float mean = s / fmaxf(cnt, 1.f);
    float var = (sq - cnt * mean * mean) / fmaxf(cnt - 1.f, 1.f);
    float stdv = sqrtf(fmaxf((cnt > 1.f) ? var : 0.f, 0.f));
    float vals[4] = {mean, stdv, cxn, cyn};
    st[which] = vals[which] * gate;
  }
}

// -------- head: h = stats @ depth_w + b, layernorm --------------------------
__global__ void __launch_bounds__(256) k_head(const float* __restrict__ stats,
                                              const float* __restrict__ dw,
                                              const float* __restrict__ db,
                                              const float* __restrict__ gamma,
                                              const float* __restrict__ beta,
                                              float* __restrict__ out) {
  int b = blockIdx.x, m = blockIdx.y;
  __shared__ float s[28];
  __shared__ float red[32];
  if (threadIdx.x < 28) s[threadIdx.x] = stats[((size_t)(m * BB + b)) * 28 + threadIdx.x];
  __syncthreads();
  float h[4];
  float sum = 0.f;
  #pragma unroll
  for (int k = 0; k < 4; ++k) {
    int d = threadIdx.x + k * 256;
    float acc = db[d];
    #pragma unroll
    for (int j = 0; j < 28; ++j) acc += s[j] * dw[j * DD + d];
    h[k] = acc;
    sum += acc;
  }
  float mu = blockSum(sum, red) * (1.0f / DD);
  float vs = 0.f;
  #pragma unroll
  for (int k = 0; k < 4; ++k) { float dlt = h[k] - mu; vs += dlt * dlt; }
  float var = blockSum(vs, red) * (1.0f / DD);
  float inv = rsqrtf(var + 1e-5f);
  float* o = out + (size_t)RGB_ELEMS;            // dep follows rgb in d_out
  #pragma unroll
  for (int k = 0; k < 4; ++k) {
    int d = threadIdx.x + k * 256;
    o[((size_t)(m * BB + b)) * DD + d] = (h[k] - mu) * inv * gamma[d] + beta[d];
  }
}

extern "C" void kernel_launch(void* const* d_in, const int* in_sizes, int n_in,
                              void* d_out, int out_size, void* d_ws, size_t ws_size,
                              hipStream_t stream) {
  const float* vt    = (const float*)d_in[0];
  const float* depth = (const float*)d_in[1];
  const int*   masks = (const int*)d_in[2];
  const float* gw    = (const float*)d_in[3];
  const float* dw    = (const float*)d_in[4];
  const float* db    = (const float*)d_in[5];
  const float* gamma = (const float*)d_in[6];
  const float* beta  = (const float*)d_in[7];
  float* out = (float*)d_out;

  char* ws = (char*)d_ws;
  size_t off = 0;
  auto alloc = [&](size_t bytes) -> char* {
    char* p = ws + off;
    off = (off + bytes + 255) & ~(size_t)255;
    return p;
  };
  float*         scores  = (float*)alloc((size_t)BB * NN * 4);
  int*           cellsum = (int*)alloc((size_t)MM * HV * HV * 4);
  float*         soft2d  = (float*)alloc((size_t)MM * HV * HV * 4);
  float*         logsoft = (float*)alloc((size_t)MM * HV * HV * 4);
  float*         scal    = (float*)alloc((size_t)MM * 8 * 4);
  float*         sumd    = (float*)alloc((size_t)MM * BB * 4);
  float*         sumd2   = (float*)alloc((size_t)MM * BB * 4);
  unsigned char* packed  = (unsigned char*)alloc((size_t)NPIX);
  float*         dpart   = (float*)alloc((size_t)BB * DGRID * 16 * 4);
  float*         wpad    = (float*)alloc((size_t)BB * NN * 16 * 4);
  float*         partial = (float*)alloc((size_t)NCHUNK * RGB_ELEMS * 4);
  float*         stats   = (float*)alloc((size_t)MM * BB * 28 * 4);

  // 1. token scores (pass 1 over visual_tokens, mask-independent)
  k_scores<<<dim3((BB * NN) / 8), dim3(256), 0, stream>>>(vt, gw, scores);
  // 2. coverage cell sums + bit-packed masks (single read of masks)
  k_cov<<<dim3(HV * HV), dim3(256), 0, stream>>>(masks, cellsum, packed);
  // 3. soft2d / log_soft / count / centroid
  k_soft<<<dim3(MM), dim3(256), 0, stream>>>(cellsum, soft2d, logsoft, scal);
  // 4. depth Σd, Σd² for all 8 masks in one pass (deterministic 2-stage)
  k_depth<<<dim3(DGRID, BB), dim3(256), 0, stream>>>(depth, packed, dpart);
  k_depth2<<<dim3(BB), dim3(256), 0, stream>>>(dpart, sumd, sumd2);
  // 5. softmax -> padded WMMA-A weights
  k_weights<<<dim3(BB, MM), dim3(256), 0, stream>>>(scores, logsoft, wpad);
  // 6. rgb = weights x tokens via f32 WMMA (pass 2 over visual_tokens, L2-hot)
  k_rgb<<<dim3(DD / 128, BB, NCHUNK), dim3(256), 0, stream>>>(vt, wpad, partial);
  k_rgbsum<<<dim3(RGB_ELEMS / 256), dim3(256), 0, stream>>>(partial, out);
  // 7. ray sampling + stats vector
  k_rays<<<dim3(BB, MM), dim3(512), 0, stream>>>(depth, soft2d, scal, sumd, sumd2, stats);
  // 8. head GEMV + layernorm
  k_head<<<dim3(BB, MM), dim3(256), 0, stream>>>(stats, dw, db, gamma, beta, out);
}